// GKO_GNN_Model_71906342469698
// MI455X (gfx1250) — compile-verified
//
#include <hip/hip_runtime.h>
#include <hip/hip_bf16.h>
#include <math.h>

// ---------------------------------------------------------------------------
// GNN forward for MI455X (gfx1250, wave32).
// Big GEMMs (K@z latent, 5x GAT node projections) run on V_WMMA_F32_16X16X4_F32
// with the exp-kernel construction fused into the A-fragment load so the 51MB
// geodesic matrix is streamed exactly once. Edge attention uses 3 passes with
// float atomics; the 205MB z_e edge tensor is never materialized.
// Workspace requirement: ~70 MB.
// ---------------------------------------------------------------------------

#define NN 50000
#define NE 800000
#define NS 256
#define DM 8.5f

typedef __attribute__((ext_vector_type(2))) float v2f;
typedef __attribute__((ext_vector_type(8))) float v8f;

__device__ __forceinline__ float sigmoidf_(float x) { return 1.0f / (1.0f + __expf(-x)); }
__device__ __forceinline__ float lrelu_(float x, float s) { return x >= 0.0f ? x : s * x; }

// float atomic max via signed/unsigned int ordering trick (works with -inf init)
__device__ __forceinline__ void atomicMaxF(float* addr, float v) {
  if (v >= 0.0f) atomicMax((int*)addr, __float_as_int(v));
  else           atomicMin((unsigned int*)addr, __float_as_uint(v));
}

// ---------------- encoder: z_sensors (256 x 64) ----------------
__global__ void encoder_kernel(const float* __restrict__ xb,
                               const float* __restrict__ Win, const float* __restrict__ bin,
                               const float* __restrict__ bWh, const float* __restrict__ tau,
                               const float* __restrict__ Wout, const float* __restrict__ bout,
                               float* __restrict__ z) {
  __shared__ float hn[4][64];
  int sl = threadIdx.x >> 6;          // sensor within block (4 per block)
  int j  = threadIdx.x & 63;          // feature
  int s  = blockIdx.x * 4 + sl;
  float tl = xb[s * 32 + 31];         // temp_seq[:, -1] == x_branch col 31
  hn[sl][j] = tanhf(tl * Win[j] + bin[j] + bWh[j]) / tau[j];
  __syncthreads();
  float acc = bout[j];
#pragma unroll
  for (int k = 0; k < 64; ++k) acc += hn[sl][k] * Wout[k * 64 + j];
  z[s * 64 + j] = tanhf(acc);
}

// ---------------- gate MLP (3 -> H1 -> H2 -> 1) ----------------
template <int H1, int H2>
__device__ float gate_mlp3(const float* xin,
                           const float* W1, const float* b1,
                           const float* g,  const float* bt,
                           const float* W2, const float* b2,
                           const float* W3, const float* b3) {
  float h1[H1];
#pragma unroll
  for (int jj = 0; jj < H1; ++jj)
    h1[jj] = xin[0] * W1[jj] + xin[1] * W1[H1 + jj] + xin[2] * W1[2 * H1 + jj] + b1[jj];
  float m = 0.0f;
#pragma unroll
  for (int jj = 0; jj < H1; ++jj) m += h1[jj];
  m *= (1.0f / H1);
  float v = 0.0f;
#pragma unroll
  for (int jj = 0; jj < H1; ++jj) { float d = h1[jj] - m; v += d * d; }
  v *= (1.0f / H1);
  float inv = rsqrtf(v + 1e-5f);
#pragma unroll
  for (int jj = 0; jj < H1; ++jj) h1[jj] = tanhf(g[jj] * (h1[jj] - m) * inv + bt[jj]);
  float h2[H2];
#pragma unroll
  for (int q = 0; q < H2; ++q) {
    float a = b2[q];
#pragma unroll
    for (int jj = 0; jj < H1; ++jj) a += h1[jj] * W2[jj * H2 + q];
    h2[q] = tanhf(a);
  }
  float o = b3[0];
#pragma unroll
  for (int q = 0; q < H2; ++q) o += h2[q] * W3[q];
  return o;
}

// h71 row layout (stride 72): [0..2]=xyz_w [3..5]=type_w [6]=ini*w [7..70]=latent [71]=0 pad
__global__ void node_feat_kernel(const float* __restrict__ x, const float* __restrict__ ini,
                                 const float* __restrict__ wscale,
                                 const float* xW1, const float* xb1, const float* xg,
                                 const float* xbt, const float* xW2, const float* xb2,
                                 const float* xW3, const float* xb3,
                                 const float* tW1, const float* tb1, const float* tg,
                                 const float* tbt, const float* tW2, const float* tb2,
                                 const float* tW3, const float* tb3,
                                 float* __restrict__ h71, int N) {
  int n = blockIdx.x * blockDim.x + threadIdx.x;
  if (n >= N) return;
  float xyz[3] = { x[n * 7 + 0], x[n * 7 + 1], x[n * 7 + 2] };
  float typ[3] = { x[n * 7 + 4], x[n * 7 + 5], x[n * 7 + 6] };
  float gx = gate_mlp3<32, 16>(xyz, xW1, xb1, xg, xbt, xW2, xb2, xW3, xb3);
  float gt = gate_mlp3<16, 8>(typ, tW1, tb1, tg, tbt, tW2, tb2, tW3, tb3);
  float* row = h71 + (size_t)n * 72;
  row[0] = gx * xyz[0]; row[1] = gx * xyz[1]; row[2] = gx * xyz[2];
  row[3] = gt * typ[0]; row[4] = gt * typ[1]; row[5] = gt * typ[2];
  row[6] = ini[n] * wscale[0];
  row[71] = 0.0f;   // pad column so layer-0 projection can use Kpad=72
}

// ---------------- latent GEMM: xl = K(50000x256) @ z(256x64), fused exp kernel
__global__ __launch_bounds__(256) void latent_wmma_kernel(
    const float* __restrict__ geo, const float* __restrict__ z,
    const float* __restrict__ ksp, float* __restrict__ xl, int M) {
  int wave = threadIdx.x >> 5;
  int lane = threadIdx.x & 31;
  int kh   = lane >> 4;        // lane half -> K pair select
  int l15  = lane & 15;
  int r0   = blockIdx.x * 128 + wave * 16;
  int anode = r0 + l15;
  bool arow = anode < M;
  float inv_ks = 1.0f / (ksp[0] + 1e-6f);
  v8f acc0 = {}, acc1 = {}, acc2 = {}, acc3 = {};
  for (int k0 = 0; k0 < NS; k0 += 4) {
    int kb = k0 + 2 * kh;
    v2f a; a[0] = 0.0f; a[1] = 0.0f;
    if (arow) {
      float d0 = geo[anode * NS + kb];
      float d1 = geo[anode * NS + kb + 1];
      a[0] = (d0 > DM) ? 0.0f : __expf(-d0 * inv_ks);
      a[1] = (d1 > DM) ? 0.0f : __expf(-d1 * inv_ks);
    }
    const float* zr0 = z + kb * 64;        // K = kb row of B
    const float* zr1 = z + (kb + 1) * 64;  // K = kb+1 row of B
    v2f b0, b1, b2, b3;
    b0[0] = zr0[ 0 + l15]; b0[1] = zr1[ 0 + l15];
    b1[0] = zr0[16 + l15]; b1[1] = zr1[16 + l15];
    b2[0] = zr0[32 + l15]; b2[1] = zr1[32 + l15];
    b3[0] = zr0[48 + l15]; b3[1] = zr1[48 + l15];
    acc0 = __builtin_amdgcn_wmma_f32_16x16x4_f32(false, a, false, b0, (short)0, acc0, false, false);
    acc1 = __builtin_amdgcn_wmma_f32_16x16x4_f32(false, a, false, b1, (short)0, acc1, false, false);
    acc2 = __builtin_amdgcn_wmma_f32_16x16x4_f32(false, a, false, b2, (short)0, acc2, false, false);
    acc3 = __builtin_amdgcn_wmma_f32_16x16x4_f32(false, a, false, b3, (short)0, acc3, false, false);
  }
#pragma unroll
  for (int r = 0; r < 8; ++r) {
    int mnode = r0 + r + 8 * kh;           // C: VGPR r holds rows r / r+8 by lane half
    if (mnode < M) {
      float* o = xl + (size_t)mnode * 64;
      o[ 0 + l15] = acc0[r];
      o[16 + l15] = acc1[r];
      o[32 + l15] = acc2[r];
      o[48 + l15] = acc3[r];
    }
  }
}

// ---------------- generic WMMA projection: out = A(MxKpad,lda) @ W(Kreal x 64) + bias
__global__ __launch_bounds__(256) void proj_wmma_kernel(
    const float* __restrict__ A, int lda, int Kreal, int Kpad,
    const float* __restrict__ W, const float* __restrict__ bias,
    float* __restrict__ out, int M) {
  int wave = threadIdx.x >> 5;
  int lane = threadIdx.x & 31;
  int kh   = lane >> 4;
  int l15  = lane & 15;
  int r0   = blockIdx.x * 128 + wave * 16;
  int anode = r0 + l15;
  bool arow = anode < M;
  v8f acc0 = {}, acc1 = {}, acc2 = {}, acc3 = {};
  for (int k0 = 0; k0 < Kpad; k0 += 4) {
    int kb = k0 + 2 * kh;
    v2f a; a[0] = 0.0f; a[1] = 0.0f;
    if (arow) {
      const float* ap = A + (size_t)anode * lda + kb;   // A padded to Kpad with zeros
      a[0] = ap[0]; a[1] = ap[1];
    }
    v2f b0, b1, b2, b3;
    float z0 = 0.0f, z1 = 0.0f, z2 = 0.0f, z3 = 0.0f;
    float y0 = 0.0f, y1 = 0.0f, y2 = 0.0f, y3 = 0.0f;
    if (kb < Kreal) {
      const float* wr = W + kb * 64;
      z0 = wr[0 + l15]; z1 = wr[16 + l15]; z2 = wr[32 + l15]; z3 = wr[48 + l15];
    }
    if (kb + 1 < Kreal) {
      const float* wr = W + (kb + 1) * 64;
      y0 = wr[0 + l15]; y1 = wr[16 + l15]; y2 = wr[32 + l15]; y3 = wr[48 + l15];
    }
    b0[0] = z0; b0[1] = y0;  b1[0] = z1; b1[1] = y1;
    b2[0] = z2; b2[1] = y2;  b3[0] = z3; b3[1] = y3;
    acc0 = __builtin_amdgcn_wmma_f32_16x16x4_f32(false, a, false, b0, (short)0, acc0, false, false);
    acc1 = __builtin_amdgcn_wmma_f32_16x16x4_f32(false, a, false, b1, (short)0, acc1, false, false);
    acc2 = __builtin_amdgcn_wmma_f32_16x16x4_f32(false, a, false, b2, (short)0, acc2, false, false);
    acc3 = __builtin_amdgcn_wmma_f32_16x16x4_f32(false, a, false, b3, (short)0, acc3, false, false);
  }
  float bb0 = bias[ 0 + l15], bb1 = bias[16 + l15];
  float bb2 = bias[32 + l15], bb3 = bias[48 + l15];
#pragma unroll
  for (int r = 0; r < 8; ++r) {
    int mnode = r0 + r + 8 * kh;
    if (mnode < M) {
      float* o = out + (size_t)mnode * 64;
      o[ 0 + l15] = acc0[r] + bb0;
      o[16 + l15] = acc1[r] + bb1;
      o[32 + l15] = acc2[r] + bb2;
      o[48 + l15] = acc3[r] + bb3;
    }
  }
}

// ---------------- latent LayerNorm -> h71 columns 7..70 ----------------
__global__ void latent_ln_kernel(const float* __restrict__ xl,
                                 const float* __restrict__ g, const float* __restrict__ b,
                                 float* __restrict__ h71, int N) {
  int n = blockIdx.x * blockDim.x + threadIdx.x;
  if (n >= N) return;
  const float* r = xl + (size_t)n * 64;
  float s = 0.0f;
#pragma unroll 8
  for (int j = 0; j < 64; ++j) s += r[j];
  float m = s * (1.0f / 64.0f);
  float v = 0.0f;
#pragma unroll 8
  for (int j = 0; j < 64; ++j) { float d = r[j] - m; v += d * d; }
  v *= (1.0f / 64.0f);
  float inv = rsqrtf(v + 1e-5f);
  float* o = h71 + (size_t)n * 72 + 7;
#pragma unroll 8
  for (int j = 0; j < 64; ++j) o[j] = g[j] * (r[j] - m) * inv + b[j];
}

// ---------------- per-layer init ----------------
__global__ void layer_init_kernel(float* accum, float* den, float* m,
                                  double* gs, double* gq) {
  int idx = blockIdx.x * blockDim.x + threadIdx.x;
  if (idx < NN * 64) accum[idx] = 0.0f;
  if (idx < NN) { den[idx] = 0.0f; m[idx] = -__builtin_inff(); }
  if (idx < 64) { gs[idx] = 0.0; gq[idx] = 0.0; }
}

// ---------------- edge pass 1: alpha + segment max ----------------
__global__ void edge_alpha_kernel(const int* __restrict__ ei, const float* __restrict__ ea,
                                  const float* __restrict__ hp,
                                  const float* __restrict__ We, const float* __restrict__ be,
                                  const float* __restrict__ attn,
                                  float* __restrict__ alphaw, float* __restrict__ mbuf, int E) {
  int t = blockIdx.x * blockDim.x + threadIdx.x;
  if (t >= E) return;
  int src = ei[t], dst = ei[E + t];
  float e0 = ea[t * 3 + 0], e1 = ea[t * 3 + 1], e2 = ea[t * 3 + 2];
  const float4* h4  = (const float4*)(hp + (size_t)src * 64);
  const float4* We4 = (const float4*)We;
  const float4* be4 = (const float4*)be;
  const float4* at4 = (const float4*)attn;
  float dot = 0.0f;
#pragma unroll
  for (int jj = 0; jj < 16; ++jj) {
    float4 w0 = We4[jj], w1 = We4[16 + jj], w2 = We4[32 + jj];
    float4 bb = be4[jj], at = at4[jj], hv = h4[jj];
    float p;
    p = sigmoidf_(e0 * w0.x + e1 * w1.x + e2 * w2.x + bb.x); dot += hv.x * p * at.x;
    p = sigmoidf_(e0 * w0.y + e1 * w1.y + e2 * w2.y + bb.y); dot += hv.y * p * at.y;
    p = sigmoidf_(e0 * w0.z + e1 * w1.z + e2 * w2.z + bb.z); dot += hv.z * p * at.z;
    p = sigmoidf_(e0 * w0.w + e1 * w1.w + e2 * w2.w + bb.w); dot += hv.w * p * at.w;
  }
  float a = lrelu_(dot, 0.1f);
  alphaw[t] = a;
  atomicMaxF(&mbuf[dst], a);
}

// ---------------- edge pass 2: softmax denominator ----------------
__global__ void edge_den_kernel(const int* __restrict__ ei, const float* __restrict__ alphaw,
                                const float* __restrict__ mbuf, float* __restrict__ den, int E) {
  int t = blockIdx.x * blockDim.x + threadIdx.x;
  if (t >= E) return;
  int dst = ei[E + t];
  atomicAdd(&den[dst], __expf(alphaw[t] - mbuf[dst]));
}

// ---------------- edge pass 3: weighted scatter (recompute z_e) ----------------
__global__ void edge_aggr_kernel(const int* __restrict__ ei, const float* __restrict__ ea,
                                 const float* __restrict__ hp,
                                 const float* __restrict__ We, const float* __restrict__ be,
                                 const float* __restrict__ alphaw, const float* __restrict__ mbuf,
                                 const float* __restrict__ den, float* __restrict__ accum, int E) {
  int t = blockIdx.x * blockDim.x + threadIdx.x;
  if (t >= E) return;
  int src = ei[t], dst = ei[E + t];
  float ac = __expf(alphaw[t] - mbuf[dst]) / (den[dst] + 1e-16f);
  float e0 = ea[t * 3 + 0], e1 = ea[t * 3 + 1], e2 = ea[t * 3 + 2];
  const float4* h4  = (const float4*)(hp + (size_t)src * 64);
  const float4* We4 = (const float4*)We;
  const float4* be4 = (const float4*)be;
  float* arow = accum + (size_t)dst * 64;
#pragma unroll
  for (int jj = 0; jj < 16; ++jj) {
    float4 w0 = We4[jj], w1 = We4[16 + jj], w2 = We4[32 + jj];
    float4 bb = be4[jj], hv = h4[jj];
    float p;
    p = sigmoidf_(e0 * w0.x + e1 * w1.x + e2 * w2.x + bb.x); atomicAdd(&arow[4 * jj + 0], hv.x * p * ac);
    p = sigmoidf_(e0 * w0.y + e1 * w1.y + e2 * w2.y + bb.y); atomicAdd(&arow[4 * jj + 1], hv.y * p * ac);
    p = sigmoidf_(e0 * w0.z + e1 * w1.z + e2 * w2.z + bb.z); atomicAdd(&arow[4 * jj + 2], hv.z * p * ac);
    p = sigmoidf_(e0 * w0.w + e1 * w1.w + e2 * w2.w + bb.w); atomicAdd(&arow[4 * jj + 3], hv.w * p * ac);
  }
}

// ---------------- GraphNorm stats (per-feature sums over nodes) ----------------
__global__ void gn_stats_kernel(const float* __restrict__ x, double* gs, double* gq, int N) {
  int f   = threadIdx.x & 63;
  int grp = blockIdx.x * (blockDim.x >> 6) + (threadIdx.x >> 6);
  int ng  = gridDim.x * (blockDim.x >> 6);
  double s = 0.0, q = 0.0;
  for (int n = grp; n < N; n += ng) {
    float v = x[(size_t)n * 64 + f];
    s += v; q += (double)v * v;
  }
  atomicAdd(&gs[f], s);
  atomicAdd(&gq[f], q);
}

// ---------------- GraphNorm apply + lrelu + residual ----------------
__global__ void gn_apply_kernel(const float* __restrict__ xin, float* __restrict__ h,
                                const double* __restrict__ gs, const double* __restrict__ gq,
                                const float* __restrict__ w, const float* __restrict__ b,
                                const float* __restrict__ ms, int N, int first) {
  int idx = blockIdx.x * blockDim.x + threadIdx.x;
  if (idx >= N * 64) return;
  int f = idx & 63;
  float m   = (float)(gs[f] * (1.0 / N));
  float ex2 = (float)(gq[f] * (1.0 / N));
  float msf = ms[f];
  float var = ex2 - m * m * msf * (2.0f - msf);   // E[(x - ms*m)^2]
  float xc  = xin[idx] - msf * m;
  float hn  = w[f] * xc * rsqrtf(var + 1e-5f) + b[f];
  hn = lrelu_(hn, 0.01f);
  h[idx] = first ? hn : hn + h[idx];
}

// ---------------- output head ----------------
__global__ void out_kernel(const float* __restrict__ h,
                           const float* __restrict__ Wo1, const float* __restrict__ bo1,
                           const float* __restrict__ go,  const float* __restrict__ bto,
                           const float* __restrict__ Wo2, const float* __restrict__ bo2,
                           float* __restrict__ out, int N) {
  int n = blockIdx.x * blockDim.x + threadIdx.x;
  if (n >= N) return;
  float o1[32];
#pragma unroll
  for (int q = 0; q < 32; ++q) o1[q] = bo1[q];
  const float* hr = h + (size_t)n * 64;
  for (int k = 0; k < 64; ++k) {
    float hv = hr[k];
#pragma unroll
    for (int q = 0; q < 32; ++q) o1[q] += hv * Wo1[k * 32 + q];
  }
  float m = 0.0f;
#pragma unroll
  for (int q = 0; q < 32; ++q) m += o1[q];
  m *= (1.0f / 32.0f);
  float v = 0.0f;
#pragma unroll
  for (int q = 0; q < 32; ++q) { float d = o1[q] - m; v += d * d; }
  v *= (1.0f / 32.0f);
  float inv = rsqrtf(v + 1e-5f);
  float o = bo2[0];
#pragma unroll
  for (int q = 0; q < 32; ++q) {
    float t = lrelu_(go[q] * (o1[q] - m) * inv + bto[q], 0.01f);
    o += t * Wo2[q];
  }
  out[n] = tanhf(o);
}

// ---------------------------------------------------------------------------
// d_in leaf order (setup_inputs dict insertion order, recursive):
//  0 x  1 x_branch  2 geodesic  3 ini  4 edge_index  5 edge_attr
//  6 kernel_scale  7 weights
//  8..15  wxyz: W1 b1 g bt W2 b2 W3 b3
// 16..23  wtype: W1 b1 g bt W2 b2 W3 b3
// 24..29  enc: Win bin bWh tau Wout bout
// 30..31  latent_ln: g b
// 32..56  gat[i]: Wn bn We be attn   (5 layers)
// 57..71  gn[i]: w b ms              (5 layers)
// 72..77  out: Wo1 bo1 go bto Wo2 bo2
// ---------------------------------------------------------------------------
extern "C" void kernel_launch(void* const* d_in, const int* in_sizes, int n_in,
                              void* d_out, int out_size, void* d_ws, size_t ws_size,
                              hipStream_t stream) {
  (void)in_sizes; (void)n_in; (void)out_size; (void)ws_size;
  const float* x    = (const float*)d_in[0];
  const float* xb   = (const float*)d_in[1];
  const float* geo  = (const float*)d_in[2];
  const float* ini  = (const float*)d_in[3];
  const int*   ei   = (const int*)d_in[4];
  const float* ea   = (const float*)d_in[5];
  const float* ksp  = (const float*)d_in[6];
  const float* wts  = (const float*)d_in[7];
  const float** P = (const float**)d_in;

  float* ws     = (float*)d_ws;
  float* z      = ws;                    // 16384
  float* xl     = z      + 16384;        // 3,200,000
  float* h71    = xl     + 3200000;      // 3,600,000 (stride 72)
  float* h64    = h71    + 3600000;      // 3,200,000
  float* hp     = h64    + 3200000;      // 3,200,000
  float* accum  = hp     + 3200000;      // 3,200,000
  float* alphaw = accum  + 3200000;      // 800,000
  float* mbuf   = alphaw + 800000;       // 50,000
  float* denb   = mbuf   + 50000;        // 50,000
  double* gs    = (double*)(denb + 50000); // 64 doubles (8B aligned)
  double* gq    = gs + 64;

  encoder_kernel<<<64, 256, 0, stream>>>(xb, P[24], P[25], P[26], P[27], P[28], P[29], z);

  node_feat_kernel<<<(NN + 255) / 256, 256, 0, stream>>>(
      x, ini, wts,
      P[8], P[9], P[10], P[11], P[12], P[13], P[14], P[15],
      P[16], P[17], P[18], P[19], P[20], P[21], P[22], P[23],
      h71, NN);

  latent_wmma_kernel<<<(NN + 127) / 128, 256, 0, stream>>>(geo, z, ksp, xl, NN);
  latent_ln_kernel<<<(NN + 255) / 256, 256, 0, stream>>>(xl, P[30], P[31], h71, NN);

  for (int i = 0; i < 5; ++i) {
    const float* Wn   = P[32 + 5 * i + 0];
    const float* bn   = P[32 + 5 * i + 1];
    const float* We   = P[32 + 5 * i + 2];
    const float* be   = P[32 + 5 * i + 3];
    const float* attn = P[32 + 5 * i + 4];
    const float* gw   = P[57 + 3 * i + 0];
    const float* gb   = P[57 + 3 * i + 1];
    const float* gms  = P[57 + 3 * i + 2];
    const float* A = (i == 0) ? h71 : h64;
    int lda  = (i == 0) ? 72 : 64;
    int Kr   = (i == 0) ? 71 : 64;
    int Kp   = (i == 0) ? 72 : 64;

    layer_init_kernel<<<(NN * 64 + 255) / 256, 256, 0, stream>>>(accum, denb, mbuf, gs, gq);
    proj_wmma_kernel<<<(NN + 127) / 128, 256, 0, stream>>>(A, lda, Kr, Kp, Wn, bn, hp, NN);
    edge_alpha_kernel<<<(NE + 255) / 256, 256, 0, stream>>>(ei, ea, hp, We, be, attn, alphaw, mbuf, NE);
    edge_den_kernel<<<(NE + 255) / 256, 256, 0, stream>>>(ei, alphaw, mbuf, denb, NE);
    edge_aggr_kernel<<<(NE + 255) / 256, 256, 0, stream>>>(ei, ea, hp, We, be, alphaw, mbuf, denb, accum, NE);
    gn_stats_kernel<<<120, 256, 0, stream>>>(accum, gs, gq, NN);
    gn_apply_kernel<<<(NN * 64 + 255) / 256, 256, 0, stream>>>(accum, h64, gs, gq, gw, gb, gms, NN, i == 0 ? 1 : 0);
  }

  out_kernel<<<(NN + 255) / 256, 256, 0, stream>>>(h64, P[72], P[73], P[74], P[75], P[76], P[77],
                                                   (float*)d_out, NN);
}